// Predictor_67104569033152
// MI455X (gfx1250) — compile-verified
//
#include <hip/hip_runtime.h>
#include <math.h>

#define ALPHA 0.5f
#define BETA  0.5f
#define D 64

typedef float v2f __attribute__((ext_vector_type(2)));
typedef float v8f __attribute__((ext_vector_type(8)));

// ---------------- degree accumulation -------------------------------------
__global__ void degree_kernel(const int* __restrict__ src, const int* __restrict__ dst,
                              float* __restrict__ deg_out, float* __restrict__ deg_in, int E) {
    int e = blockIdx.x * blockDim.x + threadIdx.x;
    if (e < E) {
        atomicAdd(&deg_out[src[e]], 1.0f);
        atomicAdd(&deg_in[dst[e]], 1.0f);
    }
}

// ---------------- n_unique + sum-of-squares of degrees --------------------
__global__ void stats_kernel(const float* __restrict__ deg_out,
                             const float* __restrict__ deg_in,
                             float* __restrict__ stats, int N) {
    int n = blockIdx.x * blockDim.x + threadIdx.x;
    float cnt = 0.f, sso = 0.f, ssi = 0.f;
    if (n < N) {
        float a = deg_out[n], b = deg_in[n];
        cnt = (a + b > 0.f) ? 1.f : 0.f;   // node present in edge_index
        sso = a * a;
        ssi = b * b;
    }
    // wave32 butterfly reduction
    for (int off = 16; off > 0; off >>= 1) {
        cnt += __shfl_xor(cnt, off);
        sso += __shfl_xor(sso, off);
        ssi += __shfl_xor(ssi, off);
    }
    if ((threadIdx.x & 31) == 0) {
        atomicAdd(&stats[0], cnt);
        atomicAdd(&stats[1], sso);
        atomicAdd(&stats[2], ssi);
    }
}

// ---------------- per-node scalar terms A[n], B[n] ------------------------
__global__ void node_scalar_kernel(const float* __restrict__ z_in,
                                   const float* __restrict__ z_out,
                                   const float* __restrict__ b_in,
                                   const float* __restrict__ b_out,
                                   const float* __restrict__ deg_out,
                                   const float* __restrict__ deg_in,
                                   const float* __restrict__ stats,
                                   float* __restrict__ Anode,
                                   float* __restrict__ Bnode, int N) {
    int n = blockIdx.x * blockDim.x + threadIdx.x;
    if (n >= N) return;
    float nu    = stats[0];
    float denom = nu - 1.0f;
    // ||deg/denom|| = sqrt(ssq)/|denom| ; reference clamps norm at 1e-12
    float nrm_o = fmaxf(sqrtf(stats[1]) / fabsf(denom), 1e-12f);
    float nrm_i = fmaxf(sqrtf(stats[2]) / fabsf(denom), 1e-12f);
    float dn_o  = (deg_out[n] / denom) / nrm_o;
    float dn_i  = (deg_in[n]  / denom) / nrm_i;
    const float* zo = z_out + (size_t)n * (D + 1);
    const float* zi = z_in  + (size_t)n * (D + 1);
    float dbo = 0.f, dbi = 0.f;
    #pragma unroll 8
    for (int j = 0; j < D; ++j) {
        dbo += zo[j] * b_out[j];
        dbi += zi[j] * b_in[j];
    }
    Anode[n] = BETA * (dn_o + zo[D]) + ALPHA * dbo;   // belongs to src side
    Bnode[n] = BETA * (dn_i + zi[D]) + ALPHA * dbi;   // belongs to dst side
}

// ---------------- WMMA f32 GEMM: Out[N x 64] = X[N x 64] * B ---------------
// TRANS_B == false : B[k][j] = W[k*64 + j]      (u_out = z_out[:, :64] @ W_out)
// TRANS_B == true  : B[k][j] = W[j*64 + k]      (q     = z_self @ W_in^T)
// One wave computes a 16-row x 64-col tile via 4 N-tiles, K unrolled by 4
// with V_WMMA_F32_16X16X4_F32 (16 k-steps -> 64 wmma ops per wave).
template <bool TRANS_B>
__global__ void wmma_gemm_kernel(const float* __restrict__ X, int ldx,
                                 const float* __restrict__ W,
                                 float* __restrict__ Out, int N) {
    const int lane  = threadIdx.x & 31;
    const int wave  = threadIdx.x >> 5;
    const int chunk = blockIdx.x * (blockDim.x >> 5) + wave;
    const int row0  = chunk * 16;
    if (row0 >= N) return;                 // uniform whole-wave exit (EXEC stays all-1s)

    const int m = lane & 15;
    int row = row0 + m;
    if (row >= N) row = N - 1;             // clamp for ragged tail (data-only, no divergence)

    const int khalf   = (lane >> 4) << 1;  // lanes 0-15 -> K {k0,k0+1}; lanes 16-31 -> {k0+2,k0+3}
    const int colbase = lane & 15;

    v8f acc[4];
    #pragma unroll
    for (int t = 0; t < 4; ++t) acc[t] = {};

    #pragma unroll
    for (int k0 = 0; k0 < D; k0 += 4) {
        const int kk = k0 + khalf;
        v2f a;                              // A 16x4 tile, ISA layout
        a.x = X[(size_t)row * ldx + kk];
        a.y = X[(size_t)row * ldx + kk + 1];
        #pragma unroll
        for (int t = 0; t < 4; ++t) {
            const int col = t * 16 + colbase;
            v2f b;                          // B 4x16 tile, K pattern mirrors A
            if (TRANS_B) {
                b.x = W[col * D + kk];
                b.y = W[col * D + kk + 1];
            } else {
                b.x = W[kk * D + col];
                b.y = W[(kk + 1) * D + col];
            }
            acc[t] = __builtin_amdgcn_wmma_f32_16x16x4_f32(
                false, a, false, b, (short)0, acc[t], false, false);
        }
    }

    // C/D layout: VGPR v -> M = v (lanes 0-15) or v+8 (lanes 16-31), N = lane&15
    const int rbase = row0 + ((lane >> 4) << 3);
    #pragma unroll
    for (int t = 0; t < 4; ++t) {
        #pragma unroll
        for (int v = 0; v < 8; ++v) {
            int r = rbase + v;
            if (r < N) Out[(size_t)r * D + t * 16 + colbase] = acc[t][v];
        }
    }
}

// ---------------- per-edge gather + dot + sigmoid (one wave per edge) ------
__global__ void edge_kernel(const int* __restrict__ src, const int* __restrict__ dst,
                            const float* __restrict__ u_out, const float* __restrict__ q,
                            const float* __restrict__ z_self, const float* __restrict__ z_in,
                            const float* __restrict__ Anode, const float* __restrict__ Bnode,
                            float* __restrict__ out, int E) {
    const int lane = threadIdx.x & 31;
    const int e    = blockIdx.x * (blockDim.x >> 5) + (threadIdx.x >> 5);
    if (e >= E) return;
    const int s = src[e];
    const int t = dst[e];
    // lane covers elements {2*lane, 2*lane+1}; coalesced 256B gathers (L2-resident)
    const float2 us = ((const float2*)(u_out  + (size_t)s * D))[lane];
    const float2 zs = ((const float2*)(z_self + (size_t)t * D))[lane];
    const float2 qs = ((const float2*)(q      + (size_t)s * D))[lane];
    const float* zi = z_in + (size_t)t * (D + 1) + 2 * lane;   // stride 65 -> scalar loads
    float p = us.x * zs.x + us.y * zs.y + qs.x * zi[0] + qs.y * zi[1];
    for (int off = 16; off > 0; off >>= 1) p += __shfl_xor(p, off);
    if (lane == 0) {
        float v = Anode[s] + Bnode[t] + ALPHA * p;
        out[e]  = 1.0f / (1.0f + expf(-v));
    }
}

extern "C" void kernel_launch(void* const* d_in, const int* in_sizes, int n_in,
                              void* d_out, int out_size, void* d_ws, size_t ws_size,
                              hipStream_t stream) {
    const float* z_in   = (const float*)d_in[0];
    const float* z_out  = (const float*)d_in[1];
    const float* z_self = (const float*)d_in[2];
    const float* W_in   = (const float*)d_in[3];
    const float* b_in   = (const float*)d_in[4];
    const float* W_out  = (const float*)d_in[5];
    const float* b_out  = (const float*)d_in[6];
    const int*   edge   = (const int*)d_in[7];

    const int N = in_sizes[2] / D;     // z_self is [N, 64]
    const int E = in_sizes[7] / 2;     // edge_index is [2, E]
    const int* src = edge;
    const int* dst = edge + E;

    // workspace layout (floats): deg_out[N] deg_in[N] stats[4] A[N] B[N] u_out[N*D] q[N*D]
    float* ws      = (float*)d_ws;
    float* deg_out = ws;
    float* deg_in  = ws + (size_t)N;
    float* stats   = ws + 2 * (size_t)N;
    float* Anode   = stats + 4;
    float* Bnode   = Anode + (size_t)N;
    float* u_out   = Bnode + (size_t)N;
    float* q       = u_out + (size_t)N * D;

    // zero the accumulated regions every call (graph-capture safe, deterministic)
    hipMemsetAsync(deg_out, 0, sizeof(float) * (2 * (size_t)N + 4), stream);

    degree_kernel<<<(E + 255) / 256, 256, 0, stream>>>(src, dst, deg_out, deg_in, E);
    stats_kernel<<<(N + 255) / 256, 256, 0, stream>>>(deg_out, deg_in, stats, N);
    node_scalar_kernel<<<(N + 255) / 256, 256, 0, stream>>>(
        z_in, z_out, b_in, b_out, deg_out, deg_in, stats, Anode, Bnode, N);

    const int chunks = (N + 15) / 16;
    const int wpb    = 8;                         // 8 waves (256 threads) per block
    dim3 gemm_grid((chunks + wpb - 1) / wpb);
    wmma_gemm_kernel<false><<<gemm_grid, wpb * 32, 0, stream>>>(z_out,  D + 1, W_out, u_out, N);
    wmma_gemm_kernel<true ><<<gemm_grid, wpb * 32, 0, stream>>>(z_self, D,     W_in,  q,     N);

    const int epb = 256 / 32;                     // 8 edges per block (wave per edge)
    edge_kernel<<<(E + epb - 1) / epb, 256, 0, stream>>>(
        src, dst, u_out, q, z_self, z_in, Anode, Bnode, (float*)d_out, E);
}